// CenterLoss_78219944395380
// MI455X (gfx1250) — compile-verified
//
#include <hip/hip_runtime.h>
#include <hip/hip_bf16.h>

typedef float v2f __attribute__((ext_vector_type(2)));
typedef float v8f __attribute__((ext_vector_type(8)));

#define BLOCKS 512
#define THREADS 256
#define WAVES_PER_BLOCK (THREADS / 32)

// Pass 1: one wave per batch row (strided). 32 lanes x float4 = 512B coalesced
// per load; center row (1KB) is contiguous so the label-gather is coalesced
// per row. Wave reduction is done through V_WMMA_F32_16X16X4_F32 with an
// all-ones B matrix: sum of all D entries == 16 * sum of all A entries,
// independent of the exact lane<->element mapping, so it is numerically exact
// f32 regardless of the VGPR layout details.
__global__ __launch_bounds__(THREADS) void center_loss_partial(
    const float* __restrict__ feat, const int* __restrict__ labels,
    const float* __restrict__ centers, float* __restrict__ partials,
    int batch, int dim4 /* = feature_dim / 4 */) {
  const int lane = threadIdx.x & 31;
  const int waveInBlock = threadIdx.x >> 5;
  const int waveGlobal = blockIdx.x * WAVES_PER_BLOCK + waveInBlock;
  const int totalWaves = gridDim.x * WAVES_PER_BLOCK;

  float acc = 0.0f;

  for (int row = waveGlobal; row < batch; row += totalWaves) {
    const int lab = labels[row];
    const float4* __restrict__ frow =
        (const float4*)(feat + (size_t)row * (size_t)(dim4 * 4));
    const float4* __restrict__ crow =
        (const float4*)(centers + (size_t)lab * (size_t)(dim4 * 4));

    // Prefetch next iteration's (random) center row to hide gather latency.
    const int nrow = row + totalWaves;
    if (nrow < batch) {
      const int nlab = labels[nrow];
      const float4* __restrict__ ncrow =
          (const float4*)(centers + (size_t)nlab * (size_t)(dim4 * 4));
      __builtin_prefetch((const void*)(ncrow + lane), 0, 1);
    }

    for (int c4 = lane; c4 < dim4; c4 += 32) {
      float4 f = frow[c4];
      float4 c = crow[c4];
      float d;
      d = f.x - c.x; acc = fmaf(d, d, acc);
      d = f.y - c.y; acc = fmaf(d, d, acc);
      d = f.z - c.z; acc = fmaf(d, d, acc);
      d = f.w - c.w; acc = fmaf(d, d, acc);
    }
  }

  // ---- Wave-level reduction via WMMA (EXEC is all-ones here: no divergence
  // has occurred in this kernel before this point). ----
  v2f a;  a[0] = acc;  a[1] = 0.0f;      // A holds the 32 lane accumulators
  v2f b;  b[0] = 1.0f; b[1] = 1.0f;      // B = all ones (4x16)
  v8f cm = {};                            // C = 0
  // D[m,n] = rowsum_m(A)  =>  sum(all D entries) = 16 * sum(all A entries)
  cm = __builtin_amdgcn_wmma_f32_16x16x4_f32(
      /*neg_a=*/false, a, /*neg_b=*/false, b,
      /*c_mod=*/(short)0, cm, /*reuse_a=*/false, /*reuse_b=*/false);

  float s = cm[0] + cm[1] + cm[2] + cm[3] + cm[4] + cm[5] + cm[6] + cm[7];
  #pragma unroll
  for (int off = 16; off > 0; off >>= 1)
    s += __shfl_xor(s, off, 32);
  s *= (1.0f / 16.0f);                    // undo the 16x replication; s = wave sum

  __shared__ float wpart[WAVES_PER_BLOCK];
  if (lane == 0) wpart[waveInBlock] = s;
  __syncthreads();

  if (threadIdx.x == 0) {
    float t = 0.0f;
    #pragma unroll
    for (int i = 0; i < WAVES_PER_BLOCK; ++i) t += wpart[i];
    partials[blockIdx.x] = t;
  }
}

// Pass 2: single-block deterministic tree reduction of the block partials.
__global__ __launch_bounds__(THREADS) void center_loss_finish(
    const float* __restrict__ partials, int n, float* __restrict__ out,
    float scale) {
  float s = 0.0f;
  for (int i = threadIdx.x; i < n; i += THREADS) s += partials[i];

  __shared__ float lds[THREADS];
  lds[threadIdx.x] = s;
  __syncthreads();
  #pragma unroll
  for (int off = THREADS / 2; off > 0; off >>= 1) {
    if (threadIdx.x < off) lds[threadIdx.x] += lds[threadIdx.x + off];
    __syncthreads();
  }
  if (threadIdx.x == 0) out[0] = lds[0] * scale;
}

extern "C" void kernel_launch(void* const* d_in, const int* in_sizes, int n_in,
                              void* d_out, int out_size, void* d_ws, size_t ws_size,
                              hipStream_t stream) {
  const float* feat    = (const float*)d_in[0];   // [B, D] f32
  const int*   labels  = (const int*)d_in[1];     // [B] int
  const float* centers = (const float*)d_in[2];   // [C, D] f32

  const int batch = in_sizes[1];                  // 16384
  const int dim   = in_sizes[0] / batch;          // 256
  const int dim4  = dim / 4;

  float* partials = (float*)d_ws;                 // BLOCKS floats (2 KB)
  float* out = (float*)d_out;

  center_loss_partial<<<BLOCKS, THREADS, 0, stream>>>(
      feat, labels, centers, partials, batch, dim4);

  const float scale = 1.0f / ((float)batch * 2.0f);
  center_loss_finish<<<1, THREADS, 0, stream>>>(partials, BLOCKS, out, scale);
}